// CVAttention_41480794144919
// MI455X (gfx1250) — compile-verified
//
#include <hip/hip_runtime.h>
#include <math.h>

// Problem constants (from reference): B=16, N=128, D=64, M=4 modes, C=6, L=2
#define NPAT 2048          // B*N states
#define DPAT 64            // patch dim
#define SDIM 1296          // 6^4 amplitudes per state
#define NLAY 2
#define NTHR 192           // 6 waves per block

typedef float v2f __attribute__((ext_vector_type(2)));
typedef float v8f __attribute__((ext_vector_type(8)));

// ---------------- workspace layout (floats) ----------------
// All Gaussian gates are REAL matrices (generators are real antisymmetric).
#define OFF_BS 0                        // [(l*2+set)*3+m] * 1296   (36x36 row-major)
#define OFF_SQ 15552                    // [(l*4+m)] * 36           (6x6)
#define OFF_DP 15840                    // [(l*4+m)] * 36
#define OFF_DE 16128                    // [(p*4+m)] * 36           per-state encoding disp

#define GT_BS 0
#define GT_SQ 1
#define GT_DP 2

template <int M1> __device__ __forceinline__ int mstride_c() {
  return (M1 == 0) ? 216 : (M1 == 1) ? 36 : (M1 == 2) ? 6 : 1;
}

// state index for 2-mode gate on modes (m,m+1): rest r in [0,36), pair k in [0,36)
template <int M2> __device__ __forceinline__ int idx2c(int r, int k) {
  if (M2 == 0) return k * 36 + r;                        // pair strides (216,36), rest (6,1)
  if (M2 == 1) return (r / 6) * 216 + (r % 6) + 6 * k;   // pair strides (36,6), rest (216,1)
  return r * 36 + k;                                     // pair strides (6,1), rest (216,36)
}

// ================= gate builder: expm of real antisymmetric generators =================
// grid: 12 BS gates + 8 SQ + 8 DISP + 8192 per-state encoding displacement gates
__global__ __launch_bounds__(64) void build_gates(
    const float* __restrict__ patches, const float* __restrict__ enc_w,
    const float* __restrict__ enc_b, const float* __restrict__ bs1,
    const float* __restrict__ sq_r, const float* __restrict__ bs2,
    const float* __restrict__ disp, float* __restrict__ ws)
{
  __shared__ float E[1296], T[1296], T2[1296];
  __shared__ float redb[64];
  const int tid = threadIdx.x;
  const int bid = blockIdx.x;

  int type, n;
  float theta, bound;
  float* outp;
  if (bid < 12) {                                   // beamsplitters: expm(theta*Gb), 36x36
    int l = bid / 6, s = (bid / 3) % 2, m = bid % 3;
    theta = (s ? bs2 : bs1)[l * 3 + m];
    type = GT_BS; n = 36; bound = 12.0f;
    outp = ws + OFF_BS + ((l * 2 + s) * 3 + m) * 1296;
  } else if (bid < 20) {                            // squeezing: expm(0.5*theta*(a^2-ad^2))
    int g = bid - 12;
    theta = sq_r[g];
    type = GT_SQ; n = 6; bound = 6.0f;
    outp = ws + OFF_SQ + g * 36;
  } else if (bid < 28) {                            // layer displacement: expm(theta*(ad-a))
    int g = bid - 20;
    theta = disp[g];
    type = GT_DP; n = 6; bound = 5.0f;
    outp = ws + OFF_DP + g * 36;
  } else {                                          // encoding displacement, theta = alpha[p,m]
    int g = bid - 28;
    int p = g >> 2, m = g & 3;
    redb[tid] = patches[p * DPAT + tid] * enc_w[m * DPAT + tid];
    __syncthreads();
    for (int off = 32; off > 0; off >>= 1) {
      if (tid < off) redb[tid] += redb[tid + off];
      __syncthreads();
    }
    theta = redb[0] + enc_b[m];
    type = GT_DP; n = 6; bound = 5.0f;
    outp = ws + OFF_DE + g * 36;
  }

  const int n2 = n * n;
  float an = fabsf(theta) * bound;
  float t = theta;
  int kpow = 0;
  while (an > 0.5f && kpow < 30) { an *= 0.5f; t *= 0.5f; kpow++; }
  for (int e = tid; e < n2; e += 64) {
    float iv = ((e % (n + 1)) == 0) ? 1.0f : 0.0f;
    E[e] = iv; T[e] = iv;
  }
  __syncthreads();
  for (int it = 1; it <= 12; ++it) {                // Taylor with sparse generator
    float sc = t / (float)it;
    for (int e = tid; e < n2; e += 64) {
      int i = e / n, c = e % n;
      float v = 0.0f;
      if (type == GT_BS) {
        int i1 = i / 6, i2 = i % 6;
        if (i1 > 0 && i2 < 5) v += sqrtf((float)(i1 * (i2 + 1))) * T[((i1 - 1) * 6 + (i2 + 1)) * 36 + c];
        if (i1 < 5 && i2 > 0) v -= sqrtf((float)((i1 + 1) * i2)) * T[((i1 + 1) * 6 + (i2 - 1)) * 36 + c];
      } else if (type == GT_SQ) {
        if (i < 4) v += 0.5f * sqrtf((float)((i + 1) * (i + 2))) * T[(i + 2) * 6 + c];
        if (i > 1) v -= 0.5f * sqrtf((float)(i * (i - 1))) * T[(i - 2) * 6 + c];
      } else {
        if (i > 0) v += sqrtf((float)i) * T[(i - 1) * 6 + c];
        if (i < 5) v -= sqrtf((float)(i + 1)) * T[(i + 1) * 6 + c];
      }
      T2[e] = sc * v;
    }
    __syncthreads();
    for (int e = tid; e < n2; e += 64) { T[e] = T2[e]; E[e] += T2[e]; }
    __syncthreads();
  }
  for (int s = 0; s < kpow; ++s) {                  // squaring
    for (int e = tid; e < n2; e += 64) {
      int i = e / n, c = e % n;
      float v = 0.0f;
      for (int j = 0; j < n; ++j) v += E[i * n + j] * E[j * n + c];
      T2[e] = v;
    }
    __syncthreads();
    for (int e = tid; e < n2; e += 64) E[e] = T2[e];
    __syncthreads();
  }
  for (int e = tid; e < n2; e += 64) outp[e] = E[e];
}

// ================= per-state gate application (mode known at compile time) =============

// real 6x6 U on mode m
template <int M1>
__device__ __forceinline__ void apply1_real(
    const float* __restrict__ U,
    float*& aR, float*& aI, float*& bR, float*& bI, float* u1, int tid)
{
  if (tid < 36) u1[tid] = U[tid];
  __syncthreads();
  const int s = mstride_c<M1>();
  for (int r = tid; r < 216; r += NTHR) {
    int base = (r / s) * (6 * s) + (r % s);
    float xr[6], xi[6];
#pragma unroll
    for (int k = 0; k < 6; ++k) { xr[k] = aR[base + k * s]; xi[k] = aI[base + k * s]; }
#pragma unroll
    for (int j = 0; j < 6; ++j) {
      float yr = 0.0f, yi = 0.0f;
#pragma unroll
      for (int k = 0; k < 6; ++k) { float u = u1[j * 6 + k]; yr += u * xr[k]; yi += u * xi[k]; }
      bR[base + j * s] = yr; bI[base + j * s] = yi;
    }
  }
  __syncthreads();
  float* t;
  t = aR; aR = bR; bR = t;
  t = aI; aI = bI; bI = t;
}

// real 36x36 U on mode pair (m,m+1) via fp32 WMMA.
// out[r,j] = sum_k S[r,k]*U[j,k]:  D(48x48) = Apad(48x36) * B(36x48), B[k,j] = Upad[j*36+k].
// 6 waves: wave = (ri, mt); each wave keeps 3 accumulators (N-tiles) and reuses A frags.
template <int M2>
__device__ __forceinline__ void apply2_wmma(
    const float* __restrict__ U,
    float*& aR, float*& aI, float*& bR, float*& bI,
    float* ApR, float* ApI, float* Up, int tid, int lane, int wave)
{
  for (int e = tid; e < 1296; e += NTHR) Up[e] = U[e];   // rows 36-47 pre-zeroed at init
  for (int e = tid; e < 1296; e += NTHR) {               // stage state, padded contiguous
    int r = e / 36, k = e - r * 36;
    int si = idx2c<M2>(r, k);
    ApR[e] = aR[si];
    ApI[e] = aI[si];
  }
  __syncthreads();

  const int mt = wave % 3;            // M tile (rest rows)
  const int ri = wave / 3;            // 0 = Re, 1 = Im
  const float* Ap = ri ? ApI : ApR;
  const int arow = mt * 16 + (lane & 15);
  const int koff = (lane >> 4) << 1;  // A/B frag: lanes 16-31 hold K+2
  const int col0 = lane & 15;

  v8f acc[3];
#pragma unroll
  for (int nt = 0; nt < 3; ++nt) acc[nt] = (v8f){0.f, 0.f, 0.f, 0.f, 0.f, 0.f, 0.f, 0.f};

  for (int ks = 0; ks < 9; ++ks) {                       // K = 36 = 9 * 4, no divergence
    const int k0 = ks * 4 + koff;
    v2f av = *(const v2f*)(Ap + arow * 36 + k0);
    v2f bv0 = *(const v2f*)(Up + (col0) * 36 + k0);
    v2f bv1 = *(const v2f*)(Up + (16 + col0) * 36 + k0);
    v2f bv2 = *(const v2f*)(Up + (32 + col0) * 36 + k0);
    acc[0] = __builtin_amdgcn_wmma_f32_16x16x4_f32(false, av, false, bv0, (short)0, acc[0], false, false);
    acc[1] = __builtin_amdgcn_wmma_f32_16x16x4_f32(false, av, false, bv1, (short)0, acc[1], false, false);
    acc[2] = __builtin_amdgcn_wmma_f32_16x16x4_f32(false, av, false, bv2, (short)0, acc[2], false, false);
  }

  // scatter D back into strided state layout (only divergent part)
  float* Bo = ri ? bI : bR;
  const int rbase = mt * 16 + ((lane >> 4) << 3);        // C/D: VGPR v -> M = v (+8 upper half)
#pragma unroll
  for (int nt = 0; nt < 3; ++nt) {
    int bcol = nt * 16 + col0;
    if (bcol < 36) {
#pragma unroll
      for (int v = 0; v < 8; ++v) {
        int row = rbase + v;
        if (row < 36) Bo[idx2c<M2>(row, bcol)] = acc[nt][v];
      }
    }
  }
  __syncthreads();
  float* t;
  t = aR; aR = bR; bR = t;
  t = aI; aI = bI; bI = t;
}

// diagonal phase gate: s *= exp(i * th * n^(1 or 2)) on mode m
template <int M1, int SQP>
__device__ __forceinline__ void apply_diag(float th, float* aR, float* aI, int tid)
{
  const int s = mstride_c<M1>();
  for (int i = tid; i < SDIM; i += NTHR) {
    int nn = (i / s) % 6;
    float ang = th * (SQP ? (float)(nn * nn) : (float)nn);
    float c = cosf(ang), sn = sinf(ang);
    float xr = aR[i], xi = aI[i];
    aR[i] = xr * c - xi * sn;
    aI[i] = xr * sn + xi * c;
  }
  __syncthreads();
}

// <X_m> partial sum: X tridiagonal with sqrt(n)
template <int M1>
__device__ __forceinline__ float quad_partial(const float* aR, const float* aI, int tid)
{
  const int s = mstride_c<M1>();
  float local = 0.0f;
  for (int i = tid; i < SDIM; i += NTHR) {
    int nn = (i / s) % 6;
    float xr = 0.0f, xi = 0.0f;
    if (nn > 0) { float q = sqrtf((float)nn); xr += q * aR[i - s]; xi += q * aI[i - s]; }
    if (nn < 5) { float q = sqrtf((float)(nn + 1)); xr += q * aR[i + s]; xi += q * aI[i + s]; }
    local += aR[i] * xr + aI[i] * xi;
  }
  return local;
}

#define APPLY1_ALL(UBASE)                                              \
  apply1_real<0>((UBASE) + 0 * 36, aR, aI, bR, bI, u1, tid);           \
  apply1_real<1>((UBASE) + 1 * 36, aR, aI, bR, bI, u1, tid);           \
  apply1_real<2>((UBASE) + 2 * 36, aR, aI, bR, bI, u1, tid);           \
  apply1_real<3>((UBASE) + 3 * 36, aR, aI, bR, bI, u1, tid);

#define APPLY2_ALL(UBASE)                                                       \
  apply2_wmma<0>((UBASE) + 0 * 1296, aR, aI, bR, bI, ApR, ApI, Up, tid, lane, wave); \
  apply2_wmma<1>((UBASE) + 1 * 1296, aR, aI, bR, bI, ApR, ApI, Up, tid, lane, wave); \
  apply2_wmma<2>((UBASE) + 2 * 1296, aR, aI, bR, bI, ApR, ApI, Up, tid, lane, wave);

#define DIAG_ALL(ARR, SQP)                                \
  apply_diag<0, SQP>((ARR)[0], aR, aI, tid);              \
  apply_diag<1, SQP>((ARR)[1], aR, aI, tid);              \
  apply_diag<2, SQP>((ARR)[2], aR, aI, tid);              \
  apply_diag<3, SQP>((ARR)[3], aR, aI, tid);

// ================= main fused kernel: one workgroup per state =================
__global__ __launch_bounds__(NTHR) void cv_forward(
    const float* __restrict__ ws, const float* __restrict__ rot1,
    const float* __restrict__ rot2, const float* __restrict__ kerr,
    const float* __restrict__ ro_w, const float* __restrict__ ro_b,
    float* __restrict__ out)
{
  __shared__ __align__(16) float b0R[SDIM], b0I[SDIM], b1R[SDIM], b1I[SDIM];
  __shared__ __align__(16) float ApR[1728], ApI[1728], Up[1728];   // 48x36 padded
  __shared__ float u1[40];
  __shared__ float red[NTHR];
  __shared__ float quad[4];

  const int tid = threadIdx.x;
  const int lane = tid & 31;
  const int wave = tid >> 5;
  const int p = blockIdx.x;

  float *aR = b0R, *aI = b0I, *bR = b1R, *bI = b1I;

  for (int i = tid; i < SDIM; i += NTHR) { aR[i] = 0.0f; aI[i] = 0.0f; }
  for (int i = 1296 + tid; i < 1728; i += NTHR) { ApR[i] = 0.0f; ApI[i] = 0.0f; Up[i] = 0.0f; }
  if (tid == 0) aR[0] = 1.0f;              // vacuum |0,0,0,0>
  __syncthreads();

  // encoding displacements (per-state real 6x6 gates)
  {
    const float* de = ws + OFF_DE + p * 4 * 36;
    APPLY1_ALL(de)
  }

  for (int l = 0; l < NLAY; ++l) {
    { const float* u = ws + OFF_BS + (l * 2 + 0) * 3 * 1296; APPLY2_ALL(u) }
    { const float* r1 = rot1 + l * 4;  DIAG_ALL(r1, 0) }
    { const float* u = ws + OFF_SQ + l * 4 * 36;             APPLY1_ALL(u) }
    { const float* u = ws + OFF_BS + (l * 2 + 1) * 3 * 1296; APPLY2_ALL(u) }
    { const float* r2 = rot2 + l * 4;  DIAG_ALL(r2, 0) }
    { const float* u = ws + OFF_DP + l * 4 * 36;             APPLY1_ALL(u) }
    { const float* kr = kerr + l * 4;  DIAG_ALL(kr, 1) }
  }

  // quadrature expectations
  float qp[4];
  qp[0] = quad_partial<0>(aR, aI, tid);
  qp[1] = quad_partial<1>(aR, aI, tid);
  qp[2] = quad_partial<2>(aR, aI, tid);
  qp[3] = quad_partial<3>(aR, aI, tid);
  for (int m = 0; m < 4; ++m) {
    red[tid] = qp[m];
    __syncthreads();
    if (tid < 64) red[tid] += red[tid + 64] + red[tid + 128];
    __syncthreads();
    for (int off = 32; off > 0; off >>= 1) {
      if (tid < off) red[tid] += red[tid + off];
      __syncthreads();
    }
    if (tid == 0) quad[m] = red[0];
    __syncthreads();
  }

  // readout: out[p, d] = sum_m quad[m] * ro_w[d, m] + ro_b[d]
  if (tid < 64) {
    float o = ro_b[tid];
#pragma unroll
    for (int m = 0; m < 4; ++m) o += quad[m] * ro_w[tid * 4 + m];
    out[p * 64 + tid] = o;
  }
}

// ================= launch =================
extern "C" void kernel_launch(void* const* d_in, const int* in_sizes, int n_in,
                              void* d_out, int out_size, void* d_ws, size_t ws_size,
                              hipStream_t stream) {
  (void)in_sizes; (void)n_in; (void)out_size; (void)ws_size;
  const float* patches = (const float*)d_in[0];
  const float* enc_w   = (const float*)d_in[1];
  const float* enc_b   = (const float*)d_in[2];
  const float* bs1     = (const float*)d_in[3];
  const float* rot1    = (const float*)d_in[4];
  const float* sq_r    = (const float*)d_in[5];
  const float* bs2     = (const float*)d_in[6];
  const float* rot2    = (const float*)d_in[7];
  const float* disp    = (const float*)d_in[8];
  const float* kerr    = (const float*)d_in[9];
  const float* ro_w    = (const float*)d_in[10];
  const float* ro_b    = (const float*)d_in[11];
  float* ws  = (float*)d_ws;
  float* out = (float*)d_out;

  build_gates<<<28 + NPAT * 4, 64, 0, stream>>>(patches, enc_w, enc_b, bs1, sq_r,
                                                bs2, disp, ws);
  cv_forward<<<NPAT, NTHR, 0, stream>>>(ws, rot1, rot2, kerr, ro_w, ro_b, out);
}